// InfoNCELoss_70746701300013
// MI455X (gfx1250) — compile-verified
//
#include <hip/hip_runtime.h>
#include <hip/hip_bf16.h>
#include <math.h>

typedef float v2f __attribute__((ext_vector_type(2)));
typedef float v8f __attribute__((ext_vector_type(8)));

#define B_  64
#define T_  2048
#define DZ  512
#define DC  256
#define K_  12
#define R_  2

// ---------------------------------------------------------------------------
// Kernel 1: pred[k][b][o] = sum_d c[b, t[b], d] * Wk_w[k][o][d] + Wk_b[k][o]
// One wave per (16-wide o tile, 16-wide b tile, k).  fp32 WMMA, Kdim=DC.
// A frag (16x4 f32): lane holds A[M=lane%16][K = vgpr + 2*(lane/16)]
// B frag (4x16 f32): lane holds B[K = vgpr + 2*(lane/16)][N = lane%16]
// ---------------------------------------------------------------------------
__global__ __launch_bounds__(32) void cpc_pred_kernel(
    const float* __restrict__ c, const float* __restrict__ Wk_w,
    const float* __restrict__ Wk_b, const int* __restrict__ t,
    float* __restrict__ pred)
{
    const int o0 = blockIdx.x * 16;
    const int b0 = blockIdx.y * 16;
    const int k  = blockIdx.z;
    const int lane = threadIdx.x & 31;
    const int lm = lane & 15;
    const int lh = lane >> 4;

    const int tb = t[b0 + lm];  // anchor time for this lane's batch row
    const float* arow = c + (size_t)(b0 + lm) * T_ * DC + (size_t)tb * DC + 2 * lh;
    const float* brow = Wk_w + (size_t)k * DZ * DC + (size_t)(o0 + lm) * DC + 2 * lh;

    v8f acc = {};
#pragma unroll 4
    for (int d = 0; d < DC; d += 4) {
        v2f a  = *(const v2f*)(arow + d);
        v2f bb = *(const v2f*)(brow + d);
        acc = __builtin_amdgcn_wmma_f32_16x16x4_f32(false, a, false, bb,
                                                    (short)0, acc, false, false);
    }

    // C/D layout: VGPR i, lanes 0-15 -> M=i, lanes 16-31 -> M=i+8; N = lane%16
    const float bias = Wk_b[(size_t)k * DZ + o0 + lm];
#pragma unroll
    for (int i = 0; i < 8; ++i) {
        const int bi = b0 + i + 8 * lh;                  // batch row (M)
        pred[(size_t)k * B_ * DZ + (size_t)bi * DZ + (o0 + lm)] = acc[i] + bias;
    }
}

// ---------------------------------------------------------------------------
// Kernel 2: s[k][b][t'] = sum_d z[b][t'][d] * pred[k][b][d]
// Block = 256 threads (8 waves) per (t-tile-group, b).  pred[:,b,:] staged in
// LDS padded to 16 rows (rows 12..15 zero) so pad columns accumulate zero.
// Streams z (256 MB) once -> memory-bound; prefetch one chunk ahead.
// ---------------------------------------------------------------------------
__global__ __launch_bounds__(256) void cpc_scores_kernel(
    const float* __restrict__ z, const float* __restrict__ pred,
    float* __restrict__ s)
{
    __shared__ float pred_lds[16 * DZ];   // 32 KB
    const int b   = blockIdx.y;
    const int tid = threadIdx.x;

    for (int idx = tid; idx < 16 * DZ; idx += 256) {
        const int n = idx >> 9;            // row (k index, padded to 16)
        const int d = idx & (DZ - 1);
        pred_lds[idx] = (n < K_)
            ? pred[(size_t)n * B_ * DZ + (size_t)b * DZ + d] : 0.0f;
    }
    __syncthreads();

    const int wave = tid >> 5;
    const int lane = tid & 31;
    const int lm = lane & 15;
    const int lh = lane >> 4;
    const int t0 = (blockIdx.x * 8 + wave) * 16;   // 16-row output tile

    const float* arow = z + (size_t)b * T_ * DZ + (size_t)(t0 + lm) * DZ + 2 * lh;
    const float* brow = pred_lds + lm * DZ + 2 * lh;

    v8f acc = {};
    for (int d0 = 0; d0 < DZ; d0 += 64) {
        if (d0 + 64 < DZ) __builtin_prefetch(arow + d0 + 64, 0, 0);
#pragma unroll 4
        for (int d = d0; d < d0 + 64; d += 4) {
            v2f a  = *(const v2f*)(arow + d);       // global_load_b64
            v2f bb = *(const v2f*)(brow + d);       // ds_load_b64
            acc = __builtin_amdgcn_wmma_f32_16x16x4_f32(false, a, false, bb,
                                                        (short)0, acc, false, false);
        }
    }

    if (lm < K_) {
#pragma unroll
        for (int i = 0; i < 8; ++i) {
            const int row = t0 + i + 8 * lh;        // t' (M)
            s[(size_t)lm * B_ * T_ + (size_t)b * T_ + row] = acc[i];
        }
    }
}

// ---------------------------------------------------------------------------
// Kernel 3: per (k,b): lse over {pos, 2047 gathered negatives} per replicate,
// online logsumexp per thread + shared-mem merge, atomicAdd scaled result.
// s (6 MB) lives in L2 so the random gathers are cheap.
// ---------------------------------------------------------------------------
__global__ __launch_bounds__(256) void cpc_loss_kernel(
    const float* __restrict__ s, const int* __restrict__ t,
    const int* __restrict__ rand_idx, float* __restrict__ out)
{
    __shared__ float sm_m[256];
    __shared__ float sm_s[256];

    const int kb  = blockIdx.x;
    const int k   = kb / B_;
    const int b   = kb - k * B_;
    const int tid = threadIdx.x;

    const int e = t[b] + k + 1;                      // positive frame index
    const float* srow = s + (size_t)k * B_ * T_ + (size_t)b * T_;
    const float pos = srow[e];

    float total = 0.0f;
    for (int r = 0; r < R_; ++r) {
        // thread 0 seeds with the positive candidate
        float m    = (tid == 0) ? pos  : -INFINITY;
        float ssum = (tid == 0) ? 1.0f : 0.0f;
        const int* ridx = rand_idx + ((size_t)k * R_ + r) * (T_ - 1);
        for (int j = tid; j < T_ - 1; j += 256) {
            int idx = ridx[j];
            idx += (idx >= e) ? 1 : 0;               // skip positive frame
            const float v = srow[idx];
            if (v > m) { ssum = ssum * expf(m - v) + 1.0f; m = v; }
            else       { ssum += expf(v - m); }
        }
        sm_m[tid] = m; sm_s[tid] = ssum;
        __syncthreads();
        for (int stride = 128; stride > 0; stride >>= 1) {
            if (tid < stride) {
                const float m1 = sm_m[tid], s1 = sm_s[tid];
                const float m2 = sm_m[tid + stride], s2 = sm_s[tid + stride];
                const float mm = fmaxf(m1, m2);
                sm_m[tid] = mm;
                sm_s[tid] = s1 * expf(m1 - mm) + s2 * expf(m2 - mm);
            }
            __syncthreads();
        }
        if (tid == 0) {
            const float lse = sm_m[0] + logf(sm_s[0]);
            total += lse - pos;                      // -log_softmax[...,0]
        }
        __syncthreads();
    }

    if (tid == 0) {
        atomicAdd(out, total * (1.0f / (float)(K_ * B_ * R_)));
    }
}

// ---------------------------------------------------------------------------
extern "C" void kernel_launch(void* const* d_in, const int* in_sizes, int n_in,
                              void* d_out, int out_size, void* d_ws, size_t ws_size,
                              hipStream_t stream) {
    const float* z    = (const float*)d_in[0];   // (B,T,Dz)
    const float* c    = (const float*)d_in[1];   // (B,T,Dc)
    const float* Wk_w = (const float*)d_in[2];   // (K,Dz,Dc)
    const float* Wk_b = (const float*)d_in[3];   // (K,Dz)
    const int*   t    = (const int*)d_in[4];     // (B,)
    const int*   ridx = (const int*)d_in[5];     // (K,R,T-1)
    (void)in_sizes; (void)n_in; (void)out_size;

    float* pred = (float*)d_ws;                          // K*B*Dz = 1.5 MB
    float* s    = pred + (size_t)K_ * B_ * DZ;           // K*B*T  = 6 MB

    hipMemsetAsync(d_out, 0, sizeof(float), stream);

    cpc_pred_kernel<<<dim3(DZ / 16, B_ / 16, K_), 32, 0, stream>>>(
        c, Wk_w, Wk_b, t, pred);

    cpc_scores_kernel<<<dim3(T_ / 128, B_), 256, 0, stream>>>(z, pred, s);

    cpc_loss_kernel<<<dim3(K_ * B_), 256, 0, stream>>>(s, t, ridx, (float*)d_out);
}